// DynamicMemoryRouter_39152921870437
// MI455X (gfx1250) — compile-verified
//
#include <hip/hip_runtime.h>
#include <hip/hip_bf16.h>
#include <math.h>

#define Bb   4
#define Nn   4096
#define Dd   1024
#define Hh   16
#define Ss   512
#define DH   64
#define DFF  4096

typedef __attribute__((ext_vector_type(2))) float v2f;
typedef __attribute__((ext_vector_type(8))) float v8f;

static __device__ __forceinline__ v8f wmma_f32(v2f a, v2f b, v8f c) {
    // D = A(16x4 f32) * B(4x16 f32) + C(16x16 f32)
    return __builtin_amdgcn_wmma_f32_16x16x4_f32(false, a, false, b, (short)0, c, false, false);
}

static __device__ __forceinline__ float gelu_exact(float x) {
    return 0.5f * x * (1.0f + erff(x * 0.70710678118654752440f));
}

// ---------------------------------------------------------------- LayerNorm
__global__ void ln_kernel(const float* __restrict__ x, const float* __restrict__ g,
                          const float* __restrict__ bia, float* __restrict__ out) {
    const int row = blockIdx.x;
    const float* xr = x + (size_t)row * Dd;
    float s = 0.f, s2 = 0.f;
    for (int i = threadIdx.x; i < Dd; i += 256) { float v = xr[i]; s += v; s2 += v * v; }
    __shared__ float sh1[256], sh2[256];
    sh1[threadIdx.x] = s; sh2[threadIdx.x] = s2; __syncthreads();
    for (int off = 128; off > 0; off >>= 1) {
        if (threadIdx.x < off) { sh1[threadIdx.x] += sh1[threadIdx.x + off];
                                 sh2[threadIdx.x] += sh2[threadIdx.x + off]; }
        __syncthreads();
    }
    const float mean = sh1[0] * (1.0f / Dd);
    const float var  = sh2[0] * (1.0f / Dd) - mean * mean;
    const float rstd = rsqrtf(var + 1e-5f);
    float* orow = out + (size_t)row * Dd;
    for (int i = threadIdx.x; i < Dd; i += 256)
        orow[i] = (xr[i] - mean) * rstd * g[i] + bia[i];
}

// ------------------------------------------- scores: R[bh] = Xh(Nx64) @ Mk[h]^T
// Block = 8 waves sharing one 16-row A strip; wave w covers s-columns [w*64, w*64+64)
__global__ void gemm_scores(const float* __restrict__ Xln, const float* __restrict__ Mk,
                            float* __restrict__ R) {
    const int lane = threadIdx.x & 31, w = threadIdx.x >> 5;
    const int half = lane >> 4, lr = lane & 15;
    const int n0 = blockIdx.x * 16;                  // 256 n-tiles
    const int s0 = w * 64;                            // 8 waves cover full S=512
    const int bh = blockIdx.z, b = bh >> 4, h = bh & 15;

    const float* Ap = Xln + ((size_t)b * Nn + n0 + lr) * Dd + h * DH;
    const float* Bp = Mk + (size_t)h * Ss * DH;      // Mk[h][s][k]
    v8f c0 = {}, c1 = {}, c2 = {}, c3 = {};
#pragma unroll
    for (int k0 = 0; k0 < DH; k0 += 4) {
        const int ka = k0 + 2 * half;
        v2f a  = *(const v2f*)(Ap + ka);
        v2f b0 = *(const v2f*)(Bp + (size_t)(s0 +  0 + lr) * DH + ka);
        v2f b1 = *(const v2f*)(Bp + (size_t)(s0 + 16 + lr) * DH + ka);
        v2f b2 = *(const v2f*)(Bp + (size_t)(s0 + 32 + lr) * DH + ka);
        v2f b3 = *(const v2f*)(Bp + (size_t)(s0 + 48 + lr) * DH + ka);
        c0 = wmma_f32(a, b0, c0); c1 = wmma_f32(a, b1, c1);
        c2 = wmma_f32(a, b2, c2); c3 = wmma_f32(a, b3, c3);
    }
    float* Rt = R + (size_t)bh * Nn * Ss;
#pragma unroll
    for (int r = 0; r < 8; r++) {
        float* dst = Rt + (size_t)(n0 + r + 8 * half) * Ss + s0 + lr;
        dst[0] = c0[r]; dst[16] = c1[r]; dst[32] = c2[r]; dst[48] = c3[r];
    }
}

// ---------------- column softmax stats over N (online max/sumexp per (b,h,s))
__global__ void colstats_kernel(const float* __restrict__ R,
                                float* __restrict__ m, float* __restrict__ l) {
    const int idx = blockIdx.x * 256 + threadIdx.x;  // over B*H*S
    const int s = idx % Ss, bh = idx / Ss;
    const float* col = R + (size_t)bh * Nn * Ss + s;
    float mr = -3.4e38f, lr_ = 0.f;
    for (int n = 0; n < Nn; n++) {
        const float v = col[(size_t)n * Ss];
        if (v > mr) { lr_ = lr_ * __expf(mr - v) + 1.f; mr = v; }
        else        { lr_ += __expf(v - mr); }
    }
    m[idx] = mr; l[idx] = lr_;
}

// -------- per-row denominator: 1e-9 + sum_s exp(r-m)/l   (one wave per row)
__global__ void rowsum_kernel(const float* __restrict__ R, const float* __restrict__ m,
                              const float* __restrict__ l, float* __restrict__ rs) {
    const int gw = (blockIdx.x * 256 + threadIdx.x) >> 5;  // B*H*N waves
    const int lane = threadIdx.x & 31;
    const int bh = gw / Nn, n = gw % Nn;
    const float* row = R + ((size_t)bh * Nn + n) * Ss;
    const float* mb = m + (size_t)bh * Ss;
    const float* lb = l + (size_t)bh * Ss;
    float acc = 0.f;
    for (int s = lane; s < Ss; s += 32) acc += __expf(row[s] - mb[s]) / lb[s];
    for (int off = 16; off > 0; off >>= 1) acc += __shfl_xor(acc, off, 32);
    if (lane == 0) rs[gw] = acc + 1e-9f;
}

// ------- read: O[b,n,h*64+d] = (softmaxed R) @ Mv[h], normalization fused in
// Wave covers full DH=64 width with 4 tiles, so per-wave A is unique: keep wave->n-tile map.
__global__ void gemm_read(const float* __restrict__ R, const float* __restrict__ m,
                          const float* __restrict__ l, const float* __restrict__ rs,
                          const float* __restrict__ Mv, float* __restrict__ Obuf) {
    const int lane = threadIdx.x & 31, w = threadIdx.x >> 5;
    const int half = lane >> 4, lr = lane & 15;
    const int n0 = (blockIdx.x * 8 + w) * 16;        // 256 n-tiles
    const int bh = blockIdx.y, b = bh >> 4, h = bh & 15;

    const float* Ar = R + ((size_t)bh * Nn + n0 + lr) * Ss;
    const float* mb = m + (size_t)bh * Ss;
    const float* lb = l + (size_t)bh * Ss;
    const float* Bp = Mv + (size_t)h * Ss * DH;      // Mv[h][k][d]
    v8f c0 = {}, c1 = {}, c2 = {}, c3 = {};
    for (int k0 = 0; k0 < Ss; k0 += 4) {
        const int ka = k0 + 2 * half;
        v2f a;
        a.x = __expf(Ar[ka]     - mb[ka])     / lb[ka];
        a.y = __expf(Ar[ka + 1] - mb[ka + 1]) / lb[ka + 1];
        const float* Br0 = Bp + (size_t)ka * DH;
        const float* Br1 = Br0 + DH;
        v2f b0, b1, b2, b3;
        b0.x = Br0[ 0 + lr]; b0.y = Br1[ 0 + lr];
        b1.x = Br0[16 + lr]; b1.y = Br1[16 + lr];
        b2.x = Br0[32 + lr]; b2.y = Br1[32 + lr];
        b3.x = Br0[48 + lr]; b3.y = Br1[48 + lr];
        c0 = wmma_f32(a, b0, c0); c1 = wmma_f32(a, b1, c1);
        c2 = wmma_f32(a, b2, c2); c3 = wmma_f32(a, b3, c3);
    }
#pragma unroll
    for (int r = 0; r < 8; r++) {
        const int row = n0 + r + 8 * half;
        const float inv = 1.0f / rs[(size_t)bh * Nn + row];  // per-row renorm over S
        float* dst = Obuf + ((size_t)b * Nn + row) * Dd + h * DH + lr;
        dst[0] = c0[r] * inv; dst[16] = c1[r] * inv;
        dst[32] = c2[r] * inv; dst[48] = c3[r] * inv;
    }
}

// ----------------------------- y = F_in + O @ Wo^T  (rows B*N, cols D, K = D)
// Block = 8 waves sharing one 16-row A strip; wave w -> col-group (blockIdx.y*8+w)
__global__ void gemm_wo(const float* __restrict__ Ob, const float* __restrict__ Wo,
                        const float* __restrict__ Fin, float* __restrict__ y) {
    const int lane = threadIdx.x & 31, w = threadIdx.x >> 5;
    const int half = lane >> 4, lr = lane & 15;
    const int r0 = blockIdx.x * 16;                  // 1024 row-tiles
    const int c0 = (blockIdx.y * 8 + w) * 64;        // 16 col-groups
    const float* Ap = Ob + (size_t)(r0 + lr) * Dd;
    v8f cc0 = {}, cc1 = {}, cc2 = {}, cc3 = {};
    for (int k0 = 0; k0 < Dd; k0 += 4) {
        const int ka = k0 + 2 * half;
        v2f a  = *(const v2f*)(Ap + ka);
        v2f b0 = *(const v2f*)(Wo + (size_t)(c0 +  0 + lr) * Dd + ka);  // B[k][c]=Wo[c][k]
        v2f b1 = *(const v2f*)(Wo + (size_t)(c0 + 16 + lr) * Dd + ka);
        v2f b2 = *(const v2f*)(Wo + (size_t)(c0 + 32 + lr) * Dd + ka);
        v2f b3 = *(const v2f*)(Wo + (size_t)(c0 + 48 + lr) * Dd + ka);
        cc0 = wmma_f32(a, b0, cc0); cc1 = wmma_f32(a, b1, cc1);
        cc2 = wmma_f32(a, b2, cc2); cc3 = wmma_f32(a, b3, cc3);
    }
#pragma unroll
    for (int r = 0; r < 8; r++) {
        const size_t row = (size_t)(r0 + r + 8 * half);
        const float* fr = Fin + row * Dd + c0 + lr;
        float* dst = y + row * Dd + c0 + lr;
        dst[0]  = fr[0]  + cc0[r]; dst[16] = fr[16] + cc1[r];
        dst[32] = fr[32] + cc2[r]; dst[48] = fr[48] + cc3[r];
    }
}

// -------------------------- H = GELU(t @ W1 + b1)  (rows B*N, cols DFF, K = D)
__global__ void gemm_ffn1(const float* __restrict__ t, const float* __restrict__ W1,
                          const float* __restrict__ b1, float* __restrict__ Hb) {
    const int lane = threadIdx.x & 31, w = threadIdx.x >> 5;
    const int half = lane >> 4, lr = lane & 15;
    const int r0 = blockIdx.x * 16;                  // 1024 row-tiles
    const int c0 = (blockIdx.y * 8 + w) * 64;        // 64 col-groups
    const float* Ap = t + (size_t)(r0 + lr) * Dd;
    v8f cc0 = {}, cc1 = {}, cc2 = {}, cc3 = {};
    for (int k0 = 0; k0 < Dd; k0 += 4) {
        const int ka = k0 + 2 * half;
        v2f a = *(const v2f*)(Ap + ka);
        const float* Br0 = W1 + (size_t)ka * DFF + c0;
        const float* Br1 = Br0 + DFF;
        v2f b0, b1v, b2, b3;
        b0.x  = Br0[ 0 + lr]; b0.y  = Br1[ 0 + lr];
        b1v.x = Br0[16 + lr]; b1v.y = Br1[16 + lr];
        b2.x  = Br0[32 + lr]; b2.y  = Br1[32 + lr];
        b3.x  = Br0[48 + lr]; b3.y  = Br1[48 + lr];
        cc0 = wmma_f32(a, b0, cc0); cc1 = wmma_f32(a, b1v, cc1);
        cc2 = wmma_f32(a, b2, cc2); cc3 = wmma_f32(a, b3, cc3);
    }
#pragma unroll
    for (int r = 0; r < 8; r++) {
        const size_t row = (size_t)(r0 + r + 8 * half);
        float* dst = Hb + row * DFF + c0 + lr;
        dst[0]  = gelu_exact(cc0[r] + b1[c0 +  0 + lr]);
        dst[16] = gelu_exact(cc1[r] + b1[c0 + 16 + lr]);
        dst[32] = gelu_exact(cc2[r] + b1[c0 + 32 + lr]);
        dst[48] = gelu_exact(cc3[r] + b1[c0 + 48 + lr]);
    }
}

// -------------------------- out = y + H @ W2 + b2  (rows B*N, cols D, K = DFF)
__global__ void gemm_ffn2(const float* __restrict__ Hb, const float* __restrict__ W2,
                          const float* __restrict__ b2, const float* __restrict__ y,
                          float* __restrict__ out) {
    const int lane = threadIdx.x & 31, w = threadIdx.x >> 5;
    const int half = lane >> 4, lr = lane & 15;
    const int r0 = blockIdx.x * 16;                  // 1024 row-tiles
    const int c0 = (blockIdx.y * 8 + w) * 64;        // 16 col-groups
    const float* Ap = Hb + (size_t)(r0 + lr) * DFF;
    v8f cc0 = {}, cc1 = {}, cc2 = {}, cc3 = {};
    for (int k0 = 0; k0 < DFF; k0 += 4) {
        const int ka = k0 + 2 * half;
        v2f a = *(const v2f*)(Ap + ka);
        const float* Br0 = W2 + (size_t)ka * Dd + c0;
        const float* Br1 = Br0 + Dd;
        v2f b0, b1v, b2, b3;
        b0.x  = Br0[ 0 + lr]; b0.y  = Br1[ 0 + lr];
        b1v.x = Br0[16 + lr]; b1v.y = Br1[16 + lr];
        b2.x  = Br0[32 + lr]; b2.y  = Br1[32 + lr];
        b3.x  = Br0[48 + lr]; b3.y  = Br1[48 + lr];
        cc0 = wmma_f32(a, b0, cc0); cc1 = wmma_f32(a, b1v, cc1);
        cc2 = wmma_f32(a, b2, cc2); cc3 = wmma_f32(a, b3, cc3);
    }
#pragma unroll
    for (int r = 0; r < 8; r++) {
        const size_t row = (size_t)(r0 + r + 8 * half);
        const float* yr = y + row * Dd + c0 + lr;
        float* dst = out + row * Dd + c0 + lr;
        dst[0]  = yr[0]  + cc0[r] + b2[c0 +  0 + lr];
        dst[16] = yr[16] + cc1[r] + b2[c0 + 16 + lr];
        dst[32] = yr[32] + cc2[r] + b2[c0 + 32 + lr];
        dst[48] = yr[48] + cc3[r] + b2[c0 + 48 + lr];
    }
}

extern "C" void kernel_launch(void* const* d_in, const int* in_sizes, int n_in,
                              void* d_out, int out_size, void* d_ws, size_t ws_size,
                              hipStream_t stream) {
    const float* F_in = (const float*)d_in[0];
    const float* Mk   = (const float*)d_in[1];
    const float* Mv   = (const float*)d_in[2];
    const float* ln_g = (const float*)d_in[3];
    const float* ln_b = (const float*)d_in[4];
    const float* Wo   = (const float*)d_in[5];
    const float* ln2g = (const float*)d_in[6];
    const float* ln2b = (const float*)d_in[7];
    const float* W1   = (const float*)d_in[8];
    const float* b1   = (const float*)d_in[9];
    const float* W2   = (const float*)d_in[10];
    const float* b2   = (const float*)d_in[11];
    float* out = (float*)d_out;

    // workspace layout (floats); Xln region reused for LN(y), R region for H
    float* ws   = (float*)d_ws;
    float* Xln  = ws;                                   // B*N*D      = 16,777,216
    float* Rbuf = Xln  + (size_t)Bb * Nn * Dd;          // B*H*N*S    = 134,217,728
    float* mBuf = Rbuf + (size_t)Bb * Hh * Nn * Ss;     // B*H*S      = 32,768
    float* lBuf = mBuf + (size_t)Bb * Hh * Ss;          // B*H*S
    float* rsB  = lBuf + (size_t)Bb * Hh * Ss;          // B*H*N      = 262,144
    float* Obuf = rsB  + (size_t)Bb * Hh * Nn;          // B*N*D
    float* ybuf = Obuf + (size_t)Bb * Nn * Dd;          // B*N*D
    float* tbuf = Xln;                                   // reuse (Xln dead after scores)
    float* Hbuf = Rbuf;                                  // reuse (R dead after read GEMM)

    // 1) x = LN(F_in)
    ln_kernel<<<Bb * Nn, 256, 0, stream>>>(F_in, ln_g, ln_b, Xln);
    // 2) R = x @ Mk^T per (b,h): block covers 16 x 512 strip (waves share A rows)
    gemm_scores<<<dim3(256, 1, Bb * Hh), 256, 0, stream>>>(Xln, Mk, Rbuf);
    // 3) column softmax stats (over N)
    colstats_kernel<<<(Bb * Hh * Ss) / 256, 256, 0, stream>>>(Rbuf, mBuf, lBuf);
    // 4) per-row denominators (renorm over S)
    rowsum_kernel<<<(Bb * Hh * Nn * 32) / 256, 256, 0, stream>>>(Rbuf, mBuf, lBuf, rsB);
    // 5) O = softmax(R) @ Mv, normalization fused into A-frag build + C store
    gemm_read<<<dim3(32, Bb * Hh), 256, 0, stream>>>(Rbuf, mBuf, lBuf, rsB, Mv, Obuf);
    // 6) y = F_in + O @ Wo^T: blocks of 16 rows x 512 cols
    gemm_wo<<<dim3(1024, 2), 256, 0, stream>>>(Obuf, Wo, F_in, ybuf);
    // 7) t = LN(y)
    ln_kernel<<<Bb * Nn, 256, 0, stream>>>(ybuf, ln2g, ln2b, tbuf);
    // 8) H = GELU(t @ W1 + b1): blocks of 16 rows x 512 cols
    gemm_ffn1<<<dim3(1024, 8), 256, 0, stream>>>(tbuf, W1, b1, Hbuf);
    // 9) out = y + H @ W2 + b2
    gemm_ffn2<<<dim3(1024, 2), 256, 0, stream>>>(Hbuf, W2, b2, ybuf, out);
}